// GetNextStates_39264591020479
// MI455X (gfx1250) — compile-verified
//
#include <hip/hip_runtime.h>
#include <hip/hip_bf16.h>
#include <stdint.h>

#define DT_STEP 0.025f
#define TPB 256

// Per-block LDS staging regions (all sizes/offsets multiples of 16 bytes).
#define ST_BYTES (TPB * 13 * 4)   // 13312 : states (and reused for output)
#define AC_BYTES (TPB * 12 * 4)   // 12288 : actions
#define FP_BYTES (TPB * 12 * 4)   // 12288 : foot positions
#define FC_BYTES (TPB * 4)        //  1024 : contacts (4 bool bytes / element)
#define OFF_ST 0
#define OFF_AC (OFF_ST + ST_BYTES)
#define OFF_FP (OFF_AC + AC_BYTES)
#define OFF_FC (OFF_FP + FP_BYTES)
#define SMEM_BYTES (OFF_FC + FC_BYTES)   // 38912 B -> 8 blocks per 320KB WGP

// CDNA5 async copies. INST_OFFSET is added to BOTH the LDS address and the
// global address (CDNA5 ISA), so one LDS-base VGPR and one
// global-base VGPR pair cover all chunks of a region via immediate offsets.
#define ASYNC_IN(lds_u32, gaddr_u64, imm)                                   \
  asm volatile("global_load_async_to_lds_b128 %0, %1, off offset:" #imm     \
               :: "v"(lds_u32), "v"(gaddr_u64) : "memory")

#define ASYNC_OUT(gaddr_u64, lds_u32, imm)                                  \
  asm volatile("global_store_async_from_lds_b128 %0, %1, off offset:" #imm  \
               :: "v"(gaddr_u64), "v"(lds_u32) : "memory")

__device__ __forceinline__ void wait_async0() {
  asm volatile("s_wait_asynccnt 0x0" ::: "memory");
}

__global__ __launch_bounds__(TPB) void GetNextStates_39264591020479_kernel(
    const float* __restrict__ params,
    const float* __restrict__ states,        // (B,13)
    const float* __restrict__ actions,       // (B,12)
    const float* __restrict__ fpos,          // (B,4,3)
    const uint8_t* __restrict__ contacts,    // (B,4) bool bytes
    float* __restrict__ out) {               // (B,13)
  __shared__ __align__(16) unsigned char smem[SMEM_BYTES];
  const int tid = (int)threadIdx.x;
  const int blk = (int)blockIdx.x;
  // addrspace(3): low 32 bits of the generic address are the LDS byte offset
  const uint32_t smem_base = (uint32_t)(uintptr_t)(void*)smem;
  const uint32_t t16 = (uint32_t)tid * 16u;

  // ---- Straight-line async staging of the block's contiguous input slabs ----
  {
    const unsigned long long gST =
        (unsigned long long)(uintptr_t)((const char*)states + (size_t)blk * ST_BYTES) + t16;
    const unsigned long long gAC =
        (unsigned long long)(uintptr_t)((const char*)actions + (size_t)blk * AC_BYTES) + t16;
    const unsigned long long gFP =
        (unsigned long long)(uintptr_t)((const char*)fpos + (size_t)blk * FP_BYTES) + t16;
    const uint32_t lST = smem_base + OFF_ST + t16;
    const uint32_t lAC = smem_base + OFF_AC + t16;
    const uint32_t lFP = smem_base + OFF_FP + t16;

    ASYNC_IN(lST, gST, 0);       // states: 832 chunks = 3*256 + 64 tail
    ASYNC_IN(lST, gST, 4096);
    ASYNC_IN(lST, gST, 8192);
    ASYNC_IN(lAC, gAC, 0);       // actions: exactly 3 uniform chunks/thread
    ASYNC_IN(lAC, gAC, 4096);
    ASYNC_IN(lAC, gAC, 8192);
    ASYNC_IN(lFP, gFP, 0);       // foot positions: 3 uniform chunks/thread
    ASYNC_IN(lFP, gFP, 4096);
    ASYNC_IN(lFP, gFP, 8192);
    if (tid < 64) {
      ASYNC_IN(lST, gST, 12288); // states tail (chunks 768..831)
      const unsigned long long gFC =
          (unsigned long long)(uintptr_t)((const char*)contacts + (size_t)blk * FC_BYTES) + t16;
      ASYNC_IN(smem_base + OFF_FC + t16, gFC, 0);  // contacts: 64 chunks total
    }
  }
  wait_async0();
  __syncthreads();

  const float* sS = (const float*)(smem + OFF_ST) + tid * 13;
  const float* sA = (const float*)(smem + OFF_AC) + tid * 12;
  const float* sF = (const float*)(smem + OFF_FP) + tid * 12;
  const uint32_t fcw = ((const uint32_t*)(smem + OFF_FC))[tid];

  float s[13];
#pragma unroll
  for (int j = 0; j < 13; ++j) s[j] = sS[j];

  // ---- Uniform per-dispatch math: 1/m and symmetric 3x3 inverse inertia ----
  const float p0 = params[0], p1 = params[1], p2 = params[2], p3 = params[3],
              p4 = params[4], p5 = params[5], p6 = params[6];
  const float inv_mass = 1.0f / p0;
  const float eps = 1e-5f;                      // ref adds 1e-5 to EVERY entry
  const float Ia = p1 * p1 + eps;               // I = L L^T + 1e-5
  const float Ib = p1 * p2 + eps;
  const float Ic = p1 * p4 + eps;
  const float Id = p2 * p2 + p3 * p3 + eps;
  const float Ie = p2 * p4 + p3 * p5 + eps;
  const float If = p4 * p4 + p5 * p5 + p6 * p6 + eps;
  const float C00 = Id * If - Ie * Ie;
  const float C01 = Ic * Ie - Ib * If;
  const float C02 = Ib * Ie - Ic * Id;
  const float rdet = 1.0f / (Ia * C00 + Ib * C01 + Ic * C02);
  const float W00 = C00 * rdet, W01 = C01 * rdet, W02 = C02 * rdet;
  const float W11 = (Ia * If - Ic * Ic) * rdet;
  const float W12 = (Ib * Ic - Ia * Ie) * rdet;
  const float W22 = (Ia * Id - Ib * Ib) * rdet;

  // ---- Per-element rotation & Euler-rate map ----
  float sr, cr, sp_, cp_, sy, cy;
  sincosf(s[0], &sr, &cr);
  sincosf(s[1], &sp_, &cp_);
  sincosf(s[2], &sy, &cy);
  const float r00 = cy * cp_, r01 = cy * sp_ * sr - sy * cr, r02 = cy * sp_ * cr + sy * sr;
  const float r10 = sy * cp_, r11 = sy * sp_ * sr + cy * cr, r12 = sy * sp_ * cr - cy * sr;
  const float r20 = -sp_,     r21 = cp_ * sr,                r22 = cp_ * cr;

  // ---- Contact-gated sum of (R p_f) x a_f and sum of a_f ----
  float nx = 0.f, ny = 0.f, nz = 0.f, fx = 0.f, fy = 0.f, fz = 0.f;
#pragma unroll
  for (int f = 0; f < 4; ++f) {
    const float m = (float)((fcw >> (8 * f)) & 0x1u);   // bool byte -> 0/1
    const float px = sF[f * 3 + 0], py = sF[f * 3 + 1], pz = sF[f * 3 + 2];
    const float vx = r00 * px + r01 * py + r02 * pz;
    const float vy = r10 * px + r11 * py + r12 * pz;
    const float vz = r20 * px + r21 * py + r22 * pz;
    const float ax = sA[f * 3 + 0], ay = sA[f * 3 + 1], az = sA[f * 3 + 2];
    nx += m * (vy * az - vz * ay);
    ny += m * (vz * ax - vx * az);
    nz += m * (vx * ay - vy * ax);
    fx += m * ax; fy += m * ay; fz += m * az;
  }

  // torque = (R W R^T) n, evaluated as R*(W*(R^T n))
  const float t0 = r00 * nx + r10 * ny + r20 * nz;
  const float t1 = r01 * nx + r11 * ny + r21 * nz;
  const float t2 = r02 * nx + r12 * ny + r22 * nz;
  const float u0 = W00 * t0 + W01 * t1 + W02 * t2;
  const float u1 = W01 * t0 + W11 * t1 + W12 * t2;
  const float u2 = W02 * t0 + W12 * t1 + W22 * t2;
  const float q0 = r00 * u0 + r01 * u1 + r02 * u2;
  const float q1 = r10 * u0 + r11 * u1 + r12 * u2;
  const float q2 = r20 * u0 + r21 * u1 + r22 * u2;

  const float rcp = 1.0f / cp_;
  const float tp = sp_ * rcp;
  float o[13];
  o[0]  = s[0] + DT_STEP * ((cy * rcp) * s[6] + (sy * rcp) * s[7]);
  o[1]  = s[1] + DT_STEP * (-sy * s[6] + cy * s[7]);
  o[2]  = s[2] + DT_STEP * ((cy * tp) * s[6] + (sy * tp) * s[7] + s[8]);
  o[3]  = s[3] + DT_STEP * s[9];
  o[4]  = s[4] + DT_STEP * s[10];
  o[5]  = s[5] + DT_STEP * s[11];
  o[6]  = s[6] + DT_STEP * q0;
  o[7]  = s[7] + DT_STEP * q1;
  o[8]  = s[8] + DT_STEP * q2;
  o[9]  = s[9]  + DT_STEP * inv_mass * fx;
  o[10] = s[10] + DT_STEP * inv_mass * fy;
  o[11] = s[11] + DT_STEP * inv_mass * fz + DT_STEP * s[12];   // A[11,12] = 1
  o[12] = s[12];

  // ---- Stage output through LDS (reuse states region: each thread owns its row)
  float* sO = (float*)(smem + OFF_ST) + tid * 13;
#pragma unroll
  for (int j = 0; j < 13; ++j) sO[j] = o[j];
  __syncthreads();

  {
    const unsigned long long gO =
        (unsigned long long)(uintptr_t)((char*)out + (size_t)blk * ST_BYTES) + t16;
    const uint32_t lO = smem_base + OFF_ST + t16;
    ASYNC_OUT(gO, lO, 0);
    ASYNC_OUT(gO, lO, 4096);
    ASYNC_OUT(gO, lO, 8192);
    if (tid < 64) { ASYNC_OUT(gO, lO, 12288); }
  }
  wait_async0();
}

extern "C" void kernel_launch(void* const* d_in, const int* in_sizes, int n_in,
                              void* d_out, int out_size, void* d_ws, size_t ws_size,
                              hipStream_t stream) {
  const float*   params   = (const float*)d_in[0];
  const float*   states   = (const float*)d_in[1];
  const float*   actions  = (const float*)d_in[2];
  const float*   fpos     = (const float*)d_in[3];
  const uint8_t* contacts = (const uint8_t*)d_in[4];
  float* out = (float*)d_out;

  const int batch = in_sizes[1] / 13;      // 262144
  const int grid = batch / TPB;            // 1024 (batch is 256-divisible)
  GetNextStates_39264591020479_kernel<<<grid, TPB, 0, stream>>>(
      params, states, actions, fpos, contacts, out);
}